// NPLinearLSTM_31421980738254
// MI455X (gfx1250) — compile-verified
//
#include <hip/hip_runtime.h>

// ---------------------------------------------------------------------------
// Problem constants
// ---------------------------------------------------------------------------
namespace {
constexpr int   HD   = 512;                       // H
constexpr int   HH   = HD * HD;                   // 262144 (channel plane)
constexpr size_t ZW  = (size_t)4 * 16 * HD * HD;  // zw elements per block
constexpr size_t ZB  = (size_t)4 * 16 * HD;       // zb elements per block
}

typedef __attribute__((ext_vector_type(2))) float v2f;
typedef __attribute__((ext_vector_type(8))) float v8f;

// ---------------------------------------------------------------------------
// Pass 1: per (g,b,c) plane: row means (cm), column means (rm), full mean,
// bias mean. grid = 5*4*16 = 320 blocks of 256 threads.
// ---------------------------------------------------------------------------
struct RedArgs {
  const float* w[5];
  const float* bias[5];
  float* rm;      // [5][4][16][512]  mean over i (column means)
  float* cm;      // [5][4][16][512]  mean over j (row means)
  float* wmean;   // [5][4][16]
  float* bmean;   // [5][4][16]
};

__global__ __launch_bounds__(256) void k_reduce(RedArgs A) {
  const int id = blockIdx.x;
  const int g = id >> 6;
  const int b = (id >> 4) & 3;
  const int c = id & 15;
  const int t = threadIdx.x;
  const float* wp = A.w[g] + (size_t)(b * 16 + c) * HH;
  const int base = ((g * 4 + b) * 16 + c) * HD;

  // column sums -> rm (coalesced: consecutive threads, consecutive j)
  float acc1 = 0.f, acc2 = 0.f;
  for (int i = 0; i < HD; ++i) {
    const float* r = wp + i * HD;
    acc1 += r[t];
    acc2 += r[t + 256];
  }
  A.rm[base + t]       = acc1 * (1.f / HD);
  A.rm[base + t + 256] = acc2 * (1.f / HD);

  // row sums -> cm (thread t owns rows t and t+256, float4 streaming)
  {
    const float4* r0 = (const float4*)(wp + (size_t)t * HD);
    const float4* r1 = (const float4*)(wp + (size_t)(t + 256) * HD);
    float s0 = 0.f, s1 = 0.f;
    for (int q = 0; q < HD / 4; ++q) {
      float4 v0 = r0[q], v1 = r1[q];
      s0 += v0.x + v0.y + v0.z + v0.w;
      s1 += v1.x + v1.y + v1.z + v1.w;
    }
    A.cm[base + t]       = s0 * (1.f / HD);
    A.cm[base + t + 256] = s1 * (1.f / HD);
  }

  __shared__ float sred[256];
  sred[t] = acc1 + acc2;                 // full-plane sum
  __syncthreads();
  for (int s = 128; s > 0; s >>= 1) {
    if (t < s) sred[t] += sred[t + s];
    __syncthreads();
  }
  if (t == 0) A.wmean[(g * 4 + b) * 16 + c] = sred[0] * (1.f / (float)HH);
  __syncthreads();

  const float* bp = A.bias[g] + (b * 16 + c) * HD;
  sred[t] = bp[t] + bp[t + 256];
  __syncthreads();
  for (int s = 128; s > 0; s >>= 1) {
    if (t < s) sred[t] += sred[t + s];
    __syncthreads();
  }
  if (t == 0) A.bmean[(g * 4 + b) * 16 + c] = sred[0] * (1.f / HD);
}

// ---------------------------------------------------------------------------
// Pass 2: per block k: rc/brc 160-wide reductions + row/col 1D convs + zb.
// One launch per block; grid = 64 blocks (b,o) of 256 threads.
// rowstore[b,o,j] = conv(row_cat) + r_b
// colstore[b,o,i] = conv(col_cat) + c_b + main_b + (wb@rc_w.T + rc_b)
// zb[b,o,n]       = conv_bias(col_cat) + bias_b + (wb@brc_w.T + brc_b)
// ---------------------------------------------------------------------------
struct VecArgs {
  const float* rowsrc[6];
  const float* colsrc[6];
  int row_cnt;     // #16-channel tensors in row_cat
  int col_cnt;
  const float* r_w;   const float* r_b;
  const float* c_w;   const float* c_b;
  const float* bias_w; const float* bias_b;
  const float* rc_w;  const float* rc_b;
  const float* brc_w; const float* brc_b;
  const float* main_b;
  const float* wmean; const float* bmean;
  float* rowstore;   // [4][16][512] (this block)
  float* colstore;   // [4][16][512]
  float* zb_out;     // [4][16][512] (slice of d_out)
};

__global__ __launch_bounds__(256) void k_vec(VecArgs A) {
  const int b = blockIdx.x >> 4;
  const int o = blockIdx.x & 15;
  const int t = threadIdx.x;
  const int row_in = A.row_cnt * 16;
  const int col_in = A.col_cnt * 16;

  __shared__ float s1[256], s2[256];
  float v = 0.f;
  if (t < 160) {
    const int tt = (t < 80) ? t : t - 80;
    const float* m = (t < 80) ? A.wmean : A.bmean;
    v = m[(tt >> 4) * 64 + b * 16 + (tt & 15)];
  }
  s1[t] = (t < 160) ? v * A.rc_w[o * 160 + t] : 0.f;
  s2[t] = (t < 160) ? v * A.brc_w[o * 160 + t] : 0.f;
  __syncthreads();
  for (int s = 128; s > 0; s >>= 1) {
    if (t < s) { s1[t] += s1[t + s]; s2[t] += s2[t + s]; }
    __syncthreads();
  }
  const float rcadd  = s1[0] + A.rc_b[o];
  const float brcadd = s2[0] + A.brc_b[o];

  for (int jj = 0; jj < 2; ++jj) {
    const int j = t + jj * 256;
    float rowacc = A.r_b[o];
    for (int cc = 0; cc < row_in; ++cc)
      rowacc += A.r_w[o * row_in + cc] * A.rowsrc[cc >> 4][(b * 16 + (cc & 15)) * HD + j];

    float colacc = A.c_b[o] + A.main_b[o] + rcadd;
    float zbacc  = A.bias_b[o] + brcadd;
    for (int cc = 0; cc < col_in; ++cc) {
      const float sv = A.colsrc[cc >> 4][(b * 16 + (cc & 15)) * HD + j];
      colacc += A.c_w[o * col_in + cc] * sv;
      zbacc  += A.bias_w[o * col_in + cc] * sv;
    }
    A.rowstore[(b * 16 + o) * HD + j] = rowacc;
    A.colstore[(b * 16 + o) * HD + j] = colacc;
    A.zb_out  [(b * 16 + o) * HD + j] = zbacc;
  }
}

// ---------------------------------------------------------------------------
// Pass 3: the big one. zw[b,o,i,j] = sum_c mw[o,c]*w[b,c,i,j]
//                                    + colstore[b,o,i] + rowstore[b,o,j]
// One wave per (blk,b,i); 32 j-tiles of 16; 16x16 fp32 tile via 4 chained
// V_WMMA_F32_16X16X4_F32. grid = 5*4*64 = 1280 blocks of 256 (8 waves).
// ---------------------------------------------------------------------------
struct MainArgs {
  const float* w[5];
  const float* mw[5];
  const float* rowstore;   // base; stride ZB per block
  const float* colstore;
  float* out;              // d_out base; stride ZW+ZB per block
};

__global__ __launch_bounds__(256) void k_zw(MainArgs A) {
  const int id   = blockIdx.x;
  const int blk  = id >> 8;
  const int rem  = id & 255;
  const int b    = rem >> 6;
  const int ig   = rem & 63;
  const int wave = threadIdx.x >> 5;
  const int lane = threadIdx.x & 31;
  const int i    = ig * 8 + wave;
  const int hi   = lane >> 4;      // half-wave selector
  const int M    = lane & 15;      // row for A, column (N) for B/C/D

  // A: 16x4 fp32 per chunk. lanes0-15: v0=K0,v1=K1 ; lanes16-31: v0=K2,v1=K3
  const float* mwp = A.mw[blk];
  v2f a0, a1, a2, a3;
  {
    int c0;
    c0 =  0 + 2 * hi; a0.x = mwp[M * 16 + c0]; a0.y = mwp[M * 16 + c0 + 1];
    c0 =  4 + 2 * hi; a1.x = mwp[M * 16 + c0]; a1.y = mwp[M * 16 + c0 + 1];
    c0 =  8 + 2 * hi; a2.x = mwp[M * 16 + c0]; a2.y = mwp[M * 16 + c0 + 1];
    c0 = 12 + 2 * hi; a3.x = mwp[M * 16 + c0]; a3.y = mwp[M * 16 + c0 + 1];
  }

  // Per-row constants: colstore[b, o, i] for the 8 D rows this lane holds.
  float colc[8];
  const float* cs = A.colstore + (size_t)blk * ZB + (size_t)(b * 16) * HD + i;
  #pragma unroll
  for (int vv = 0; vv < 8; ++vv) colc[vv] = cs[(vv + 8 * hi) * HD];

  const float* rs = A.rowstore + (size_t)blk * ZB + (size_t)(b * 16) * HD;
  const float* wplane = A.w[blk] + (size_t)(b * 16) * HH + (size_t)i * HD;
  float* outp = A.out + (size_t)blk * (ZW + ZB) + (size_t)(b * 16) * HH + (size_t)i * HD;

  for (int jt = 0; jt < 32; ++jt) {
    const int jn = jt * 16 + M;           // this lane's column index j
    const float* xb = wplane + jn;

    // B: 4x16 fp32 per chunk. VGPR v holds rows v (lanes0-15) / v+2 (16-31)
    v2f b0, b1, b2, b3;
    int c0;
    c0 =  0 + 2 * hi; b0.x = xb[(size_t)c0 * HH]; b0.y = xb[(size_t)(c0 + 1) * HH];
    c0 =  4 + 2 * hi; b1.x = xb[(size_t)c0 * HH]; b1.y = xb[(size_t)(c0 + 1) * HH];
    c0 =  8 + 2 * hi; b2.x = xb[(size_t)c0 * HH]; b2.y = xb[(size_t)(c0 + 1) * HH];
    c0 = 12 + 2 * hi; b3.x = xb[(size_t)c0 * HH]; b3.y = xb[(size_t)(c0 + 1) * HH];

    // C init: broadcast terms folded into the accumulator.
    v8f cacc;
    #pragma unroll
    for (int vv = 0; vv < 8; ++vv)
      cacc[vv] = colc[vv] + rs[(vv + 8 * hi) * HD + jn];

    cacc = __builtin_amdgcn_wmma_f32_16x16x4_f32(false, a0, false, b0, (short)0, cacc, false, false);
    cacc = __builtin_amdgcn_wmma_f32_16x16x4_f32(false, a1, false, b1, (short)0, cacc, false, false);
    cacc = __builtin_amdgcn_wmma_f32_16x16x4_f32(false, a2, false, b2, (short)0, cacc, false, false);
    cacc = __builtin_amdgcn_wmma_f32_16x16x4_f32(false, a3, false, b3, (short)0, cacc, false, false);

    // D store per ISA layout: VGPR v -> rows v (lanes0-15) / v+8 (lanes16-31)
    #pragma unroll
    for (int vv = 0; vv < 8; ++vv)
      outp[(size_t)(vv + 8 * hi) * HH + jn] = cacc[vv];
  }
}

// ---------------------------------------------------------------------------
// Host launcher
// ---------------------------------------------------------------------------
extern "C" void kernel_launch(void* const* d_in, const int* in_sizes, int n_in,
                              void* d_out, int out_size, void* d_ws, size_t ws_size,
                              hipStream_t stream) {
  (void)in_sizes; (void)n_in; (void)out_size; (void)ws_size;

  const float* w[5];
  const float* bias[5];
  for (int g = 0; g < 5; ++g) {
    w[g]    = (const float*)d_in[2 * g];
    bias[g] = (const float*)d_in[2 * g + 1];
  }

  // Workspace layout (floats): rm | cm | wmean | bmean | rowstore | colstore
  float* ws      = (float*)d_ws;
  float* rm      = ws;                 // 5*4*16*512 = 163840
  float* cm      = ws + 163840;        // 163840
  float* wmean   = ws + 327680;        // 320
  float* bmean   = ws + 328000;        // 320
  float* rowst   = ws + 328320;        // 163840
  float* colst   = ws + 492160;        // 163840  (total ~2.6 MB)

  // ---- Pass 1
  RedArgs ra;
  for (int g = 0; g < 5; ++g) { ra.w[g] = w[g]; ra.bias[g] = bias[g]; }
  ra.rm = rm; ra.cm = cm; ra.wmean = wmean; ra.bmean = bmean;
  k_reduce<<<320, 256, 0, stream>>>(ra);

  // param set (i2h[0..2] then h2h[0..1]) for each output block
  const int setOf[5] = {0, 3, 1, 4, 2};   // i2h0, h2h0, i2h1, h2h1, i2h2
  // source tables: kind 0=rm, 1=cm, 2=bias ; paired with weight index g
  const int rowK[5][6] = {{0,-1,-1,-1,-1,-1},{0,1,2,1,2,-1},{0,1,2,1,2,0},{0,1,2,1,2,-1},{0,1,2,1,2,0}};
  const int rowG[5][6] = {{0, 0, 0, 0, 0, 0},{1,1,1,0,0, 0},{2,0,0,1,1,1},{3,3,3,2,2, 0},{4,2,2,3,3,3}};
  const int rowN[5]    = {1, 5, 6, 5, 6};
  const int colK[5][6] = {{1,2,1,2,0,0},{1,2,0,0,-1,-1},{1,2,1,2,0,0},{1,2,0,-1,-1,-1},{1,2,-1,-1,-1,-1}};
  const int colG[5][6] = {{0,0,1,1,1,2},{1,1,1,2, 0, 0},{2,2,3,3,3,4},{3,3,3, 0, 0, 0},{4,4, 0, 0, 0, 0}};
  const int colN[5]    = {6, 4, 6, 3, 2};

  auto src = [&](int kind, int g) -> const float* {
    if (kind == 0) return rm + g * 32768;
    if (kind == 1) return cm + g * 32768;
    return bias[g];
  };

  // ---- Pass 2 (5 launches)
  for (int k = 0; k < 5; ++k) {
    VecArgs va;
    for (int s = 0; s < 6; ++s) {
      va.rowsrc[s] = (s < rowN[k]) ? src(rowK[k][s], rowG[k][s]) : nullptr;
      va.colsrc[s] = (s < colN[k]) ? src(colK[k][s], colG[k][s]) : nullptr;
    }
    va.row_cnt = rowN[k];
    va.col_cnt = colN[k];
    const int pb = 10 + 12 * setOf[k];
    va.r_w    = (const float*)d_in[pb + 4];  va.r_b    = (const float*)d_in[pb + 5];
    va.c_w    = (const float*)d_in[pb + 6];  va.c_b    = (const float*)d_in[pb + 7];
    va.bias_w = (const float*)d_in[pb + 8];  va.bias_b = (const float*)d_in[pb + 9];
    va.rc_w   = (const float*)d_in[pb + 2];  va.rc_b   = (const float*)d_in[pb + 3];
    va.brc_w  = (const float*)d_in[pb + 10]; va.brc_b  = (const float*)d_in[pb + 11];
    va.main_b = (const float*)d_in[pb + 1];
    va.wmean = wmean; va.bmean = bmean;
    va.rowstore = rowst + (size_t)k * ZB;
    va.colstore = colst + (size_t)k * ZB;
    va.zb_out   = (float*)d_out + (size_t)k * (ZW + ZB) + ZW;
    k_vec<<<64, 256, 0, stream>>>(va);
  }

  // ---- Pass 3 (WMMA)
  MainArgs ma;
  for (int k = 0; k < 5; ++k) {
    ma.w[k]  = w[k];
    ma.mw[k] = (const float*)d_in[10 + 12 * setOf[k] + 0];
  }
  ma.rowstore = rowst;
  ma.colstore = colst;
  ma.out = (float*)d_out;
  k_zw<<<1280, 256, 0, stream>>>(ma);
}